// GCN_block_6408091205941
// MI455X (gfx1250) — compile-verified
//
#include <hip/hip_runtime.h>
#include <hip/hip_bf16.h>
#include <math.h>

// ---------------- problem constants (match reference) ----------------
#define NN      50000
#define EE      800000
#define D_IN    256
#define HEADS   8
#define CH      32
#define HC      256
#define LIN_OUT 256
#define EPS_BN  1e-5f
#define NEG_SLOPE 0.2f
#define ENT     (EE + NN)       // edges + self loops
#define XS_STRIDE 260           // 256 + 4 pad -> lanes hit distinct LDS banks (stride%64==4)

typedef __attribute__((ext_vector_type(2))) float v2f;
typedef __attribute__((ext_vector_type(8))) float v8f;

// ---------------- small helpers ----------------
__device__ __forceinline__ void atomicMaxF(float* addr, float val) {
    // sign-magnitude monotone encoding: non-neg -> int max, neg -> uint min
    if (val >= 0.0f) atomicMax((int*)addr, __float_as_int(val));
    else             atomicMin((unsigned int*)addr, __float_as_uint(val));
}

__global__ __launch_bounds__(256) void fill_f32(float* __restrict__ p, float v, int n) {
    int i = blockIdx.x * blockDim.x + threadIdx.x;
    if (i < n) p[i] = v;
}

// ---------------- self-loop attr = mean of incoming edge_attr ----------------
__global__ __launch_bounds__(256)
void edge_mean_acc(const int* __restrict__ dst, const float* __restrict__ ea,
                   float* __restrict__ cnt, float* __restrict__ easum) {
    int i = blockIdx.x * blockDim.x + threadIdx.x;
    if (i >= EE) return;
    int d = dst[i];
    atomicAdd(&cnt[d], 1.0f);
    atomicAdd(&easum[d * 3 + 0], ea[i * 3 + 0]);
    atomicAdd(&easum[d * 3 + 1], ea[i * 3 + 1]);
    atomicAdd(&easum[d * 3 + 2], ea[i * 3 + 2]);
}

__global__ __launch_bounds__(256)
void loop_attr_finalize(float* __restrict__ easum, const float* __restrict__ cnt) {
    int i = blockIdx.x * blockDim.x + threadIdx.x;     // over N*3
    if (i >= NN * 3) return;
    float c = fmaxf(cnt[i / 3], 1.0f);
    easum[i] = easum[i] / c;                           // in-place -> loop_attr
}

// ---------------- f32 WMMA GEMM with LDS-staged A panel ----------------
// Block (8 waves) owns one 16-row tile of X, staged once in LDS (16KB, padded
// stride for conflict-free lane reads). Each wave computes two adjacent 16x16
// output tiles (col tiles 2w, 2w+1) via V_WMMA_F32_16X16X4_F32, K step 4.
// A (16x4): lanes 0-15 row=lane K={k,k+1}; lanes 16-31 row=lane-16 K={k+2,k+3}.
// B (4x16): lanes 0-15 col=lane K rows {k,k+1}; lanes 16-31 col=lane-16 K rows {k+2,k+3}.
// C/D (16x16): vgpr i = row (i + 8*(lane>=16)), col = lane&15.
__global__ __launch_bounds__(256)
void gemm256_wmma_f32(const float* __restrict__ X, const float* __restrict__ W,
                      float* __restrict__ Y, int row_tiles) {
    __shared__ float xs[16 * XS_STRIDE];               // 16 rows, padded
    int row_tile = blockIdx.x;
    if (row_tile >= row_tiles) return;                 // block-uniform
    int wave = threadIdx.x >> 5;                       // 0..7
    int lane = threadIdx.x & 31;

    // cooperative stage: 16 rows x 256 cols = 1024 float4, 4 per thread (b128 ds stores)
    {
        const float4* srcp = (const float4*)(X + (size_t)row_tile * 16 * HC);
#pragma unroll
        for (int j = 0; j < 4; ++j) {
            int idx = threadIdx.x + j * 256;           // float4 index in panel
            int r = idx >> 6, c4 = idx & 63;           // row, float4-within-row
            *(float4*)(xs + r * XS_STRIDE + c4 * 4) = srcp[(size_t)r * 64 + c4];
        }
    }
    __syncthreads();

    int half  = lane >> 4;                             // 0: K+0/1, 1: K+2/3
    int l15   = lane & 15;
    int khalf = half * 2;
    const float* xrow  = xs + l15 * XS_STRIDE;         // LDS, conflict-free
    const float* wcol0 = W + (wave * 32)      + l15;   // col tile 2w
    const float* wcol1 = W + (wave * 32 + 16) + l15;   // col tile 2w+1
    v8f acc0 = {}, acc1 = {};
#pragma unroll 4
    for (int k = 0; k < 256; k += 4) {
        if ((k & 63) == 0) __builtin_prefetch(wcol0 + (size_t)(k + 64) * HC, 0, 1);
        v2f a, b0, b1;
        a.x  = xrow[k + khalf];
        a.y  = xrow[k + khalf + 1];
        b0.x = wcol0[(size_t)(k + khalf) * HC];
        b0.y = wcol0[(size_t)(k + khalf + 1) * HC];
        b1.x = wcol1[(size_t)(k + khalf) * HC];
        b1.y = wcol1[(size_t)(k + khalf + 1) * HC];
        acc0 = __builtin_amdgcn_wmma_f32_16x16x4_f32(false, a, false, b0,
                                                     (short)0, acc0, false, false);
        acc1 = __builtin_amdgcn_wmma_f32_16x16x4_f32(false, a, false, b1,
                                                     (short)0, acc1, false, false);
    }
    int c0    = wave * 32 + l15;
    int rbase = row_tile * 16 + half * 8;
#pragma unroll
    for (int i = 0; i < 8; ++i) {
        Y[(size_t)(rbase + i) * HC + c0]      = acc0[i];
        Y[(size_t)(rbase + i) * HC + c0 + 16] = acc1[i];
    }
}

// ---------------- per-(node,head) attention dots ----------------
__global__ __launch_bounds__(256)
void node_att_dots(const float* __restrict__ h, const float* __restrict__ att_src,
                   const float* __restrict__ att_dst, float* __restrict__ a_src,
                   float* __restrict__ a_dst) {
    int idx = blockIdx.x * blockDim.x + threadIdx.x;   // over N*HEADS
    if (idx >= NN * HEADS) return;
    int n = idx >> 3, hd = idx & 7;
    const float* hp = h + (size_t)n * HC + hd * CH;
    const float* as = att_src + hd * CH;
    const float* ad = att_dst + hd * CH;
    float s1 = 0.f, s2 = 0.f;
#pragma unroll
    for (int c = 0; c < CH; ++c) { float v = hp[c]; s1 += v * as[c]; s2 += v * ad[c]; }
    a_src[idx] = s1; a_dst[idx] = s2;
}

// wea[h][d] = sum_c W_edge[d, h*32+c] * att_edge[h, c]   (8x3 = 24 values)
__global__ void compute_wea(const float* __restrict__ W_edge,
                            const float* __restrict__ att_edge, float* __restrict__ wea) {
    int t = threadIdx.x;
    if (t >= HEADS * 3) return;
    int hd = t / 3, d = t % 3;
    float s = 0.f;
    for (int c = 0; c < CH; ++c) s += W_edge[d * HC + hd * CH + c] * att_edge[hd * CH + c];
    wea[t] = s;
}

// ---------------- alpha = leaky(a_src[s] + a_dst[d] + ea.wea); segment max ----------------
__global__ __launch_bounds__(256)
void alpha_kernel(const int* __restrict__ src, const int* __restrict__ dst,
                  const float* __restrict__ edge_attr, const float* __restrict__ loop_attr,
                  const float* __restrict__ a_src, const float* __restrict__ a_dst,
                  const float* __restrict__ wea, float* __restrict__ alpha,
                  float* __restrict__ amax) {
    int idx = blockIdx.x * blockDim.x + threadIdx.x;   // over ENT*HEADS
    if (idx >= ENT * HEADS) return;
    int i = idx >> 3, hd = idx & 7;
    int s, d; const float* ear;
    if (i < EE) { s = src[i]; d = dst[i]; ear = edge_attr + (size_t)i * 3; }
    else        { s = d = i - EE;         ear = loop_attr + (size_t)(i - EE) * 3; }
    const float* w = wea + hd * 3;
    float a = a_src[s * HEADS + hd] + a_dst[d * HEADS + hd]
            + ear[0] * w[0] + ear[1] * w[1] + ear[2] * w[2];
    a = (a >= 0.f) ? a : NEG_SLOPE * a;
    alpha[idx] = a;
    atomicMaxF(&amax[d * HEADS + hd], a);
}

// ---------------- expa = exp(alpha - amax[dst]); segment sum -> denom ----------------
__global__ __launch_bounds__(256)
void exp_kernel(const int* __restrict__ dst, const float* __restrict__ amax,
                float* __restrict__ alpha, float* __restrict__ denom) {
    int idx = blockIdx.x * blockDim.x + threadIdx.x;   // over ENT*HEADS
    if (idx >= ENT * HEADS) return;
    int i = idx >> 3, hd = idx & 7;
    int d = (i < EE) ? dst[i] : (i - EE);
    float e = expf(alpha[idx] - amax[d * HEADS + hd]);
    alpha[idx] = e;                                    // alpha now holds expa
    atomicAdd(&denom[d * HEADS + hd], e);
}

// ---------------- scatter: out0[dst] += (expa/denom) * h[src]  (one wave per edge) ----------------
__global__ __launch_bounds__(256)
void message_kernel(const int* __restrict__ src, const int* __restrict__ dst,
                    const float* __restrict__ expa, const float* __restrict__ denom,
                    const float* __restrict__ h, float* __restrict__ out0) {
    int gw   = (blockIdx.x * blockDim.x + threadIdx.x) >> 5;
    int lane = threadIdx.x & 31;
    if (gw >= ENT) return;
    int s, d;
    if (gw < EE) { s = src[gw]; d = dst[gw]; } else { s = d = gw - EE; }
    const float* hrow = h + (size_t)s * HC;
    float* orow       = out0 + (size_t)d * HC;
#pragma unroll
    for (int k = 0; k < HEADS; ++k) {                  // channel = k*32 + lane, head = k
        float attn = expa[(size_t)gw * HEADS + k] / denom[d * HEADS + k];
        int ch = k * 32 + lane;
        atomicAdd(&orow[ch], attn * hrow[ch]);
    }
}

// ---------------- batch-norm (training mode, biased var) ----------------
__global__ __launch_bounds__(256)
void bn_stats(const float* __restrict__ X, float* __restrict__ stats, int nrows, int chunk) {
    int ch = threadIdx.x;                               // 256 channels
    int r0 = blockIdx.x * chunk;
    int r1 = min(r0 + chunk, nrows);
    float s = 0.f, sq = 0.f;
    for (int r = r0; r < r1; ++r) { float v = X[(size_t)r * HC + ch]; s += v; sq += v * v; }
    atomicAdd(&stats[ch], s);
    atomicAdd(&stats[HC + ch], sq);
}

__global__ __launch_bounds__(256)
void bn_apply_relu(const float* __restrict__ X, const float* __restrict__ stats,
                   const float* __restrict__ gamma, const float* __restrict__ beta,
                   float* __restrict__ Y, int n_elems) {
    int idx = blockIdx.x * blockDim.x + threadIdx.x;
    if (idx >= n_elems) return;
    int ch = idx & (HC - 1);
    float invN = 1.0f / (float)NN;
    float mu  = stats[ch] * invN;
    float var = stats[HC + ch] * invN - mu * mu;
    float y = (X[idx] - mu) * rsqrtf(var + EPS_BN) * gamma[ch] + beta[ch];
    Y[idx] = fmaxf(y, 0.0f);
}

// ---------------- host launch ----------------
extern "C" void kernel_launch(void* const* d_in, const int* in_sizes, int n_in,
                              void* d_out, int out_size, void* d_ws, size_t ws_size,
                              hipStream_t stream) {
    const float* x         = (const float*)d_in[0];
    const int*   adj       = (const int*)d_in[1];     // [2,E]
    const float* edge_attr = (const float*)d_in[2];
    const float* W_src     = (const float*)d_in[3];
    const float* att_src   = (const float*)d_in[4];
    const float* att_dst   = (const float*)d_in[5];
    const float* W_edge    = (const float*)d_in[6];
    const float* att_edge  = (const float*)d_in[7];
    // d_in[8] bias, d_in[12] b_lin: cancel exactly through the following BN means.
    const float* gamma0    = (const float*)d_in[9];
    const float* beta0     = (const float*)d_in[10];
    const float* W_lin     = (const float*)d_in[11];
    const float* gamma1    = (const float*)d_in[13];
    const float* beta1     = (const float*)d_in[14];
    float*       out       = (float*)d_out;

    const int* src = adj;
    const int* dst = adj + EE;

    // workspace carve-out
    char* p = (char*)d_ws;
    auto carve = [&](size_t bytes) { char* r = p; p += (bytes + 255) & ~(size_t)255; return (float*)r; };
    float* h     = carve((size_t)NN * HC * 4);     // x @ W_src   (later reused as BN0 output)
    float* out0  = carve((size_t)NN * HC * 4);     // aggregated messages (later reused as GEMM2 out)
    float* expa  = carve((size_t)ENT * HEADS * 4); // alpha, then expa
    float* asrc  = carve((size_t)NN * HEADS * 4);
    float* adst  = carve((size_t)NN * HEADS * 4);
    float* amax  = carve((size_t)NN * HEADS * 4);
    float* denom = carve((size_t)NN * HEADS * 4);
    float* cnt   = carve((size_t)NN * 4);
    float* lattr = carve((size_t)NN * 3 * 4);      // ea_sum, then loop_attr (in place)
    float* wea   = carve(24 * 4);
    float* st0   = carve(2 * HC * 4);
    float* st1   = carve(2 * HC * 4);

    const int T = 256;
    auto blks = [](long n, int t) { return (int)((n + t - 1) / t); };

    // 1) init accumulators (must be re-zeroed every call: atomics accumulate)
    fill_f32<<<blks(NN, T), T, 0, stream>>>(cnt, 0.f, NN);
    fill_f32<<<blks(NN * 3, T), T, 0, stream>>>(lattr, 0.f, NN * 3);
    fill_f32<<<blks(NN * HEADS, T), T, 0, stream>>>(denom, 0.f, NN * HEADS);
    fill_f32<<<blks(NN * HEADS, T), T, 0, stream>>>(amax, -INFINITY, NN * HEADS);
    fill_f32<<<blks((long)NN * HC, T), T, 0, stream>>>(out0, 0.f, NN * HC);
    fill_f32<<<1, T, 0, stream>>>(st0, 0.f, 2 * HC);
    fill_f32<<<1, T, 0, stream>>>(st1, 0.f, 2 * HC);

    // 2) self-loop attr: per-dst mean of incoming edge_attr
    edge_mean_acc<<<blks(EE, T), T, 0, stream>>>(dst, edge_attr, cnt, lattr);
    loop_attr_finalize<<<blks(NN * 3, T), T, 0, stream>>>(lattr, cnt);

    // 3) h = x @ W_src   (WMMA f32, LDS-staged A; 1 block per 16-row tile)
    {
        int row_tiles = NN / 16;                   // 3125 exactly
        gemm256_wmma_f32<<<row_tiles, T, 0, stream>>>(x, W_src, h, row_tiles);
    }

    // 4) per-node attention scalars; tiny wea table
    node_att_dots<<<blks((long)NN * HEADS, T), T, 0, stream>>>(h, att_src, att_dst, asrc, adst);
    compute_wea<<<1, 32, 0, stream>>>(W_edge, att_edge, wea);

    // 5) alpha + leaky-relu + segment max
    alpha_kernel<<<blks((long)ENT * HEADS, T), T, 0, stream>>>(
        src, dst, edge_attr, lattr, asrc, adst, wea, expa, amax);

    // 6) exp + segment sum
    exp_kernel<<<blks((long)ENT * HEADS, T), T, 0, stream>>>(dst, amax, expa, denom);

    // 7) scatter messages (1 wave per edge; f32 global atomics, mostly L2-resident)
    message_kernel<<<blks((long)ENT * 32, T), T, 0, stream>>>(src, dst, expa, denom, h, out0);

    // 8) BN0 + relu  (bias cancels through BN mean) -> write into h (dead)
    {
        int chunk = 256, nb = blks(NN, chunk);
        bn_stats<<<nb, T, 0, stream>>>(out0, st0, NN, chunk);
        bn_apply_relu<<<blks((long)NN * HC, T), T, 0, stream>>>(out0, st0, gamma0, beta0, h, NN * HC);
    }

    // 9) z = y @ W_lin  (b_lin cancels through BN mean) -> write into out0 (dead)
    {
        int row_tiles = NN / 16;
        gemm256_wmma_f32<<<row_tiles, T, 0, stream>>>(h, W_lin, out0, row_tiles);
    }

    // 10) BN1 + relu -> d_out
    {
        int chunk = 256, nb = blks(NN, chunk);
        bn_stats<<<nb, T, 0, stream>>>(out0, st1, NN, chunk);
        bn_apply_relu<<<blks((long)NN * HC, T), T, 0, stream>>>(out0, st1, gamma1, beta1, out, NN * HC);
    }
    (void)in_sizes; (void)n_in; (void)out_size; (void)ws_size;
}